// RMRKDLoss_76390288327718
// MI455X (gfx1250) — compile-verified
//
#include <hip/hip_runtime.h>
#include <math.h>

#define NCLS 19
#define NCH  19
#define HWPX (512*512)
#define NIMG 8
#define CHUNK 256
#define LPITCH (CHUNK+1)            // LDS pitch (257 floats) -> conflict-free column reads
#define BLOCKS_PER_IMG 64
#define CHUNKS_PER_BLOCK ((HWPX/CHUNK)/BLOCKS_PER_IMG)   // 16

typedef __attribute__((ext_vector_type(2))) float v2f;
typedef __attribute__((ext_vector_type(8))) float v8f;

// D = A(16x4, one-hot classes) x B(4x16, channel values) + C, full f32 precision.
__device__ __forceinline__ v8f wmma4(v2f a, v2f b, v8f c) {
    return __builtin_amdgcn_wmma_f32_16x16x4_f32(
        /*neg_a=*/false, a, /*neg_b=*/false, b,
        /*c_mod=*/(short)0, c, /*reuse_a=*/false, /*reuse_b=*/false);
}

__global__ __launch_bounds__(256) void zero_ws_kernel(float* ws, int n) {
    int i = blockIdx.x * 256 + threadIdx.x;
    if (i < n) ws[i] = 0.0f;
}

// Per-class sums of S and T channels + per-class pixel counts, via one-hot WMMA.
__global__ __launch_bounds__(256) void class_sums_kernel(
    const float* __restrict__ S, const float* __restrict__ T,
    const int* __restrict__ gt,
    float* sumsS, float* sumsT, float* cnts)
{
    __shared__ float sTile[NCH * LPITCH];
    __shared__ float tTile[NCH * LPITCH];
    __shared__ int   lab[CHUNK];
    __shared__ float binsS[NCLS * 20];
    __shared__ float binsT[NCLS * 20];
    __shared__ float cloc[NCLS];

    const int tid  = threadIdx.x;
    const int n    = blockIdx.x / BLOCKS_PER_IMG;
    const int bi   = blockIdx.x % BLOCKS_PER_IMG;
    const int lane = tid & 31;
    const int wave = tid >> 5;
    const int half = lane >> 4;      // 0: lanes 0-15, 1: lanes 16-31
    const int m    = lane & 15;

    for (int i = tid; i < NCLS * 20; i += 256) { binsS[i] = 0.0f; binsT[i] = 0.0f; }
    if (tid < NCLS) cloc[tid] = 0.0f;

    // WMMA accumulators: (class-tile, channel-tile) for S (c..) and T (d..)
    v8f c00 = {}, c01 = {}, c10 = {}, c11 = {};
    v8f d00 = {}, d01 = {}, d10 = {}, d11 = {};

    const float* Sn = S + (size_t)n * NCH * HWPX;
    const float* Tn = T + (size_t)n * NCH * HWPX;
    const int*   Gn = gt + (size_t)n * HWPX;

    for (int ck = 0; ck < CHUNKS_PER_BLOCK; ++ck) {
        const int p0 = (bi + ck * BLOCKS_PER_IMG) * CHUNK;
        __syncthreads();                       // protect LDS from previous iter readers
        // Stage labels (each pixel counted exactly once per image).
        {
            int g = Gn[p0 + tid];
            lab[tid] = g;
            atomicAdd(&cloc[g], 1.0f);         // ds_add_f32
        }
        // Stage both pred tensors, coalesced per channel row.
        for (int c = 0; c < NCH; ++c) {
            sTile[c * LPITCH + tid] = Sn[(size_t)c * HWPX + p0 + tid];
            tTile[c * LPITCH + tid] = Tn[(size_t)c * HWPX + p0 + tid];
        }
        __syncthreads();

        // Each wave owns 32 pixels of the chunk -> 8 K-steps of 4 pixels.
        const int base = wave * 32;
        for (int step = 0; step < 8; ++step) {
            // A 16x4 f32 layout: VGPR j holds K = 2*half + j (ISA 7.12.2).
            // B 4x16 assumed symmetric: VGPR j, lane-half -> K = 2*half + j.
            const int q = base + step * 4 + 2 * half;   // chunk-local pixel of K-slot 0
            const int l0 = lab[q], l1 = lab[q + 1];

            v2f a0, a1;
            a0[0] = (l0 == m)      ? 1.0f : 0.0f;  a0[1] = (l1 == m)      ? 1.0f : 0.0f;
            a1[0] = (l0 == m + 16) ? 1.0f : 0.0f;  a1[1] = (l1 == m + 16) ? 1.0f : 0.0f;

            const int   c1  = m + 16;
            const float msk = (c1 < NCH) ? 1.0f : 0.0f;      // value-select, EXEC unchanged
            const int   c1c = (c1 < NCH) ? c1 : (NCH - 1);

            v2f bs0, bs1, bt0, bt1;
            bs0[0] = sTile[m * LPITCH + q];           bs0[1] = sTile[m * LPITCH + q + 1];
            bt0[0] = tTile[m * LPITCH + q];           bt0[1] = tTile[m * LPITCH + q + 1];
            bs1[0] = msk * sTile[c1c * LPITCH + q];   bs1[1] = msk * sTile[c1c * LPITCH + q + 1];
            bt1[0] = msk * tTile[c1c * LPITCH + q];   bt1[1] = msk * tTile[c1c * LPITCH + q + 1];

            c00 = wmma4(a0, bs0, c00);  c01 = wmma4(a0, bs1, c01);
            c10 = wmma4(a1, bs0, c10);  c11 = wmma4(a1, bs1, c11);
            d00 = wmma4(a0, bt0, d00);  d01 = wmma4(a0, bt1, d01);
            d10 = wmma4(a1, bt0, d10);  d11 = wmma4(a1, bt1, d11);
        }
    }
    __syncthreads();

    // C/D 16x16 f32 layout: VGPR r -> M = r + 8*half; N = lane&15 (per half).
    for (int r = 0; r < 8; ++r) {
        const int row = r + 8 * half;        // class within tile (0..15)
        const int ch0 = m, ch1 = m + 16;
        const int cls0 = row;                // always < 16 < NCLS
        const int cls1 = row + 16;
        atomicAdd(&binsS[cls0 * 20 + ch0], c00[r]);
        atomicAdd(&binsT[cls0 * 20 + ch0], d00[r]);
        if (ch1 < NCH)  { atomicAdd(&binsS[cls0 * 20 + ch1], c01[r]);
                          atomicAdd(&binsT[cls0 * 20 + ch1], d01[r]); }
        if (cls1 < NCLS){ atomicAdd(&binsS[cls1 * 20 + ch0], c10[r]);
                          atomicAdd(&binsT[cls1 * 20 + ch0], d10[r]); }
        if (cls1 < NCLS && ch1 < NCH) {
                          atomicAdd(&binsS[cls1 * 20 + ch1], c11[r]);
                          atomicAdd(&binsT[cls1 * 20 + ch1], d11[r]); }
    }
    __syncthreads();

    // Block -> global (native f32 atomics; 722 addrs per block).
    if (tid < NCLS * NCH) {
        const int k = tid / NCH, c = tid % NCH;
        unsafeAtomicAdd(&sumsS[n * NCLS * NCH + tid], binsS[k * 20 + c]);
        unsafeAtomicAdd(&sumsT[n * NCLS * NCH + tid], binsT[k * 20 + c]);
    }
    if (tid < NCLS) unsafeAtomicAdd(&cnts[n * NCLS + tid], cloc[tid]);
}

// Means, pairwise distances (+1e-6 on each component diff, as in reference), loss.
__global__ __launch_bounds__(384) void finalize_kernel(
    const float* __restrict__ sumsS, const float* __restrict__ sumsT,
    const float* __restrict__ cnts, float* out)
{
    __shared__ float vS[NCLS * NCH], vT[NCLS * NCH];
    __shared__ float tot;
    const int tid = threadIdx.x;
    if (tid == 0) tot = 0.0f;

    for (int n = 0; n < NIMG; ++n) {
        __syncthreads();
        if (tid < NCLS * NCH) {
            const int k = tid / NCH;
            const float den = cnts[n * NCLS + k] + 1e-6f;
            vS[tid] = sumsS[n * NCLS * NCH + tid] / den;
            vT[tid] = sumsT[n * NCLS * NCH + tid] / den;
        }
        __syncthreads();
        if (tid < NCLS * NCLS) {
            const int i = tid / NCLS, j = tid % NCLS;
            if (i < j) {
                float as = 0.0f, at = 0.0f;
                for (int c = 0; c < NCH; ++c) {
                    const float es = vS[i * NCH + c] - vS[j * NCH + c] + 1e-6f;
                    const float et = vT[i * NCH + c] - vT[j * NCH + c] + 1e-6f;
                    as += es * es;  at += et * et;
                }
                const float dd = sqrtf(at) - sqrtf(as);
                atomicAdd(&tot, 0.5f * dd * dd);
            }
        }
    }
    __syncthreads();
    if (tid == 0) out[0] = 0.5f * tot / (float)NIMG;
}

extern "C" void kernel_launch(void* const* d_in, const int* in_sizes, int n_in,
                              void* d_out, int out_size, void* d_ws, size_t ws_size,
                              hipStream_t stream) {
    const float* S  = (const float*)d_in[0];
    const float* T  = (const float*)d_in[1];
    const int*   gt = (const int*)d_in[2];

    float* ws    = (float*)d_ws;
    float* sumsS = ws;
    float* sumsT = ws + NIMG * NCLS * NCH;
    float* cnts  = ws + 2 * NIMG * NCLS * NCH;
    const int wsElems = 2 * NIMG * NCLS * NCH + NIMG * NCLS;

    zero_ws_kernel<<<(wsElems + 255) / 256, 256, 0, stream>>>(ws, wsElems);
    class_sums_kernel<<<NIMG * BLOCKS_PER_IMG, 256, 0, stream>>>(
        S, T, gt, sumsS, sumsT, cnts);
    finalize_kernel<<<1, 384, 0, stream>>>(sumsS, sumsT, cnts, (float*)d_out);
}